// LaserPhaseNoise_57303453663817
// MI455X (gfx1250) — compile-verified
//
#include <hip/hip_runtime.h>

typedef __attribute__((ext_vector_type(2))) float v2f;
typedef __attribute__((ext_vector_type(8))) float v8f;

// ---------------------------------------------------------------------------
// Phase 1: pn[j] = float( f16(exp(theta[j % M])) / f16(N) )
// Matches reference rounding: exp in fp32, cast fp16, divide by fp16(N).
// ---------------------------------------------------------------------------
__global__ void lpn_pn_kernel(const float* __restrict__ theta,
                              float* __restrict__ pn,
                              int M, int Ntot) {
  int j = blockIdx.x * blockDim.x + threadIdx.x;
  if (j < Ntot) {
    _Float16 h  = (_Float16)expf(theta[j % M]);
    _Float16 nh = (_Float16)(float)Ntot;
    pn[j] = (float)(_Float16)(h / nh);
  }
}

// ---------------------------------------------------------------------------
// Phase 2: out[i][j] = pn[j] * x[i] via rank-1 V_WMMA_F32_16X16X4_F32 tiles.
// One wave per 16x16 tile; 8 waves per 256-thread block (wave32).
// A (16x4): K=0 column = x[row0..row0+15]  -> VGPR0 (a.x), lanes 0-15
// B (4x16): K=0 row    = pn[col0..col0+15] -> VGPR0 (b.x), lanes 0-15
// D (16x16 f32, 8 VGPRs): lane l -> col = l%16, row = v + 8*(l>=16)
// Stores are non-temporal: 400MB stream must not thrash 192MB L2.
// ---------------------------------------------------------------------------
__global__ void lpn_outer_wmma_kernel(const float* __restrict__ x,
                                      const float* __restrict__ pn,
                                      float* __restrict__ out,
                                      int Ntot) {
  const int lane        = threadIdx.x & 31;
  const int waveInBlock = threadIdx.x >> 5;
  const int wavesPerBlk = blockDim.x >> 5;
  const int tilesPerRow = Ntot >> 4;
  const long long totalTiles = (long long)tilesPerRow * (long long)tilesPerRow;

  long long tile = (long long)blockIdx.x * wavesPerBlk + waveInBlock;
  if (tile >= totalTiles) return;   // wave-uniform: EXEC stays all-1s below

  const int tr = (int)(tile / tilesPerRow);
  const int tc = (int)(tile % tilesPerRow);
  const int row0 = tr << 4;
  const int col0 = tc << 4;

  // Rank-1 operands: only K=0 lives in VGPR0 lanes 0-15 (per 32-bit A layout).
  const int l16 = lane & 15;
  float xa = x[row0 + l16];     // all lanes load the same 64B -> L2/WGP$ hit
  float pb = pn[col0 + l16];

  v2f a; a.x = (lane < 16) ? xa : 0.0f; a.y = 0.0f;
  v2f b; b.x = (lane < 16) ? pb : 0.0f; b.y = 0.0f;
  v8f c = {};

  v8f d = __builtin_amdgcn_wmma_f32_16x16x4_f32(
      /*neg_a=*/false, a, /*neg_b=*/false, b,
      /*c_mod=*/(short)0, c, /*reuse_a=*/false, /*reuse_b=*/false);

  // Scatter accumulators: VGPR v -> rows row0+v+ (lane>=16 ? 8 : 0), col = col0+l16.
  // Each store instruction writes two contiguous 64B row segments.
  const int mHi = (lane >> 4) << 3;               // 0 or 8
  long long base = (long long)(row0 + mHi) * Ntot + (col0 + l16);
#pragma unroll
  for (int v = 0; v < 8; ++v) {
    __builtin_nontemporal_store(d[v], &out[base + (long long)v * Ntot]);
  }
}

// ---------------------------------------------------------------------------
// Scalar fallback (only used if N % 16 != 0; not hit for N = 10000).
// ---------------------------------------------------------------------------
__global__ void lpn_outer_scalar_kernel(const float* __restrict__ x,
                                        const float* __restrict__ pn,
                                        float* __restrict__ out,
                                        int Ntot) {
  long long idx = (long long)blockIdx.x * blockDim.x + threadIdx.x;
  long long total = (long long)Ntot * (long long)Ntot;
  if (idx < total) {
    int i = (int)(idx / Ntot);
    int j = (int)(idx % Ntot);
    __builtin_nontemporal_store(pn[j] * x[i], &out[idx]);
  }
}

extern "C" void kernel_launch(void* const* d_in, const int* in_sizes, int n_in,
                              void* d_out, int out_size, void* d_ws, size_t ws_size,
                              hipStream_t stream) {
  const float* x     = (const float*)d_in[0];   // input_data [N] fp32
  const float* theta = (const float*)d_in[1];   // theta_noise [M] fp32
  // d_in[2] = k_rep (scalar); implied by N/M via j % M, not needed on device.
  const int N = in_sizes[0];
  const int M = in_sizes[1];

  float* pn  = (float*)d_ws;    // N floats of scratch (40 KB)
  float* out = (float*)d_out;   // N*N fp32

  // Phase 1: build pn vector.
  lpn_pn_kernel<<<(N + 255) / 256, 256, 0, stream>>>(theta, pn, M, N);

  if ((N & 15) == 0) {
    // Phase 2: WMMA outer-product tiles, 8 waves (tiles) per block.
    const int tilesPerRow = N >> 4;
    const long long totalTiles = (long long)tilesPerRow * tilesPerRow;
    const int wavesPerBlock = 8;                      // 256 threads, wave32
    const long long blocks = (totalTiles + wavesPerBlock - 1) / wavesPerBlock;
    lpn_outer_wmma_kernel<<<dim3((unsigned)blocks), 256, 0, stream>>>(x, pn, out, N);
  } else {
    const long long total = (long long)N * N;
    const long long blocks = (total + 255) / 256;
    lpn_outer_scalar_kernel<<<dim3((unsigned)blocks), 256, 0, stream>>>(x, pn, out, N);
  }
}